// Emb_84988812853465
// MI455X (gfx1250) — compile-verified
//
#include <hip/hip_runtime.h>
#include <stdint.h>

// ---------------------------------------------------------------------------
// Config switches (flip if a path fails to assemble on this toolchain)
// ---------------------------------------------------------------------------
#define USE_ASYNC_LDS 1

typedef float v2f __attribute__((ext_vector_type(2)));
typedef float v8f __attribute__((ext_vector_type(8)));

#define KPC   12          // pieces
#define NKB   4           // king buckets
#define DOUT  128
#define PIECE_DIM (KPC * 64)   // 768
#define BATCHES_PER_BLOCK 4
#define NVB   (BATCHES_PER_BLOCK * 2)   // 8 virtual batches (mover+waiter)
#define ROWS  (NVB * 32)                // 256 gathered rows per block
#define ROW_BYTES (DOUT * 4)            // 512 B per row

// KING_BUCKETS: sq 56,57 -> 3 ; 60 -> 1 ; 62,63 -> 2 ; else 0
__device__ __forceinline__ int king_bucket(int sq) {
    if (sq == 56 || sq == 57) return 3;
    if (sq == 60) return 1;
    if (sq == 62 || sq == 63) return 2;
    return 0;
}

// ---------------------------------------------------------------------------
// Kernel 1: merged = tiles + (pieces+ranks+files)*fmask + noking
// w table: (NKB*K*64, 128) f32 into workspace
// ---------------------------------------------------------------------------
__global__ void build_table_kernel(const float* __restrict__ pieces,
                                   const float* __restrict__ ranks,
                                   const float* __restrict__ files,
                                   const float* __restrict__ noking,
                                   const float* __restrict__ tiles,
                                   const float* __restrict__ fmask,
                                   float* __restrict__ w) {
    int r = blockIdx.x;            // 0..3071 : ((kb*12+p)*8+rank)*8+file
    int d = threadIdx.x;           // 0..127
    int file = r & 7;
    int rank = (r >> 3) & 7;
    int kp   = r >> 6;             // kb*12 + p
    int p    = kp % KPC;
    float m  = fmask[r];
    float v  = tiles[(size_t)r * DOUT + d]
             + (pieces[(size_t)kp * DOUT + d]
              + ranks[(size_t)(kp * 8 + rank) * DOUT + d]
              + files[(size_t)(kp * 8 + file) * DOUT + d]) * m
             + noking[(size_t)((p * 8 + rank) * 8 + file) * DOUT + d];
    w[(size_t)r * DOUT + d] = v;
}

// ---------------------------------------------------------------------------
// Kernel 2: gather 256 rows per block into LDS via async-to-LDS copies,
// then segment-sum with chained V_WMMA_F32_16X16X4_F32 (block-diagonal
// selector A), clip, store both outputs.
// ---------------------------------------------------------------------------
__global__ __launch_bounds__(256)
void accum_kernel(const int*   __restrict__ values,
                  const int*   __restrict__ kings,
                  const float* __restrict__ wtab,
                  float*       __restrict__ out,
                  int B, int L) {
    extern __shared__ char smem[];
    float*    g_rows = (float*)smem;                        // ROWS*DOUT f32
    unsigned* rowoff = (unsigned*)(smem + ROWS * ROW_BYTES); // ROWS u32

    const int tid  = threadIdx.x;       // 0..255
    const int lane = tid & 31;
    const int wave = tid >> 5;          // 0..7
    const int bbase = blockIdx.x * BATCHES_PER_BLOCK;

    // ---- per-row gathered-row byte offsets (one row per thread) ----
    {
        int r    = tid;                 // 0..255
        int vb   = r >> 5;              // virtual batch 0..7
        int tok  = r & 31;
        int side = vb >> 2;             // 0 = mover, 1 = waiter
        int bi   = vb & 3;
        int batch = bbase + bi;
        int v = values[batch * L + tok];
        int row;
        if (side == 0) {
            int mk = kings[batch * 2 + 0];
            row = king_bucket(mk) * PIECE_DIM + v;
        } else {
            int wk  = kings[batch * 2 + 1];
            int wkf = (7 - (wk >> 3)) * 8 + (wk & 7);
            int piece = v >> 6, sq = v & 63;
            int fv = ((piece + (KPC / 2)) % KPC) * 64 + (7 - (sq >> 3)) * 8 + (sq & 7);
            row = king_bucket(wkf) * PIECE_DIM + fv;
        }
        rowoff[r] = (unsigned)row * (unsigned)ROW_BYTES;
    }
    __syncthreads();

    // ---- stage 256 rows (128 KB) into LDS ----
    // Each thread issues 32 async b128 copies: segment sg = i*256+tid moves
    // 16 B of row (sg>>5) at sub-offset (sg&31)*16.
    const unsigned ldsbase = (unsigned)__builtin_amdgcn_groupstaticsize();
#pragma unroll 4
    for (int i = 0; i < 32; ++i) {
        int sg  = i * 256 + tid;
        int row = sg >> 5;
        int seg = sg & 31;
        unsigned long long ga =
            (unsigned long long)(uintptr_t)((const char*)wtab + rowoff[row] + seg * 16);
        unsigned ldsoff = ldsbase + (unsigned)sg * 16u;
#if USE_ASYNC_LDS
        asm volatile("global_load_async_to_lds_b128 %0, %1, off"
                     :: "v"(ldsoff), "v"(ga)
                     : "memory");
#else
        const float4 dv = *(const float4*)(uintptr_t)ga;
        *(float4*)((char*)smem + (size_t)sg * 16) = dv;
#endif
    }
#if USE_ASYNC_LDS
#if __has_builtin(__builtin_amdgcn_s_wait_asynccnt)
    __builtin_amdgcn_s_wait_asynccnt(0);
#else
    asm volatile("s_wait_asynccnt 0" ::: "memory");
#endif
#endif
    __syncthreads();

    // ---- segment reduction: D[16x16] = S(block-diag ones) x G, f32 WMMA ----
    const int n0    = wave * 16;        // each wave owns a 16-wide DOUT chunk
    const int col   = n0 + (lane & 15);
    const int khalf = lane >> 4;        // lane-group half selects K slots

    float acc[8];
#if __has_builtin(__builtin_amdgcn_wmma_f32_16x16x4_f32)
    v8f c = {0.f, 0.f, 0.f, 0.f, 0.f, 0.f, 0.f, 0.f};
    for (int m0 = 0; m0 < NVB; ++m0) {
        float av = ((lane & 15) == m0) ? 1.0f : 0.0f;
        v2f a; a.x = av; a.y = av;          // A[m][k] = delta(m,m0), all k
#pragma unroll
        for (int kc = 0; kc < 8; ++kc) {    // 8 K-chunks of 4 cover 32 tokens
            int t0 = m0 * 32 + kc * 4 + khalf;   // B layout: vgpr0 -> k=0/1
            int t1 = t0 + 2;                     //           vgpr1 -> k=2/3
            v2f b;
            b.x = g_rows[t0 * DOUT + col];
            b.y = g_rows[t1 * DOUT + col];
            c = __builtin_amdgcn_wmma_f32_16x16x4_f32(
                    false, a, false, b, (short)0, c, false, false);
        }
    }
#pragma unroll
    for (int r = 0; r < 8; ++r) acc[r] = c[r];
#else
    // VALU fallback with identical D-tile layout (lanes 0-15 hold data)
#pragma unroll
    for (int r = 0; r < 8; ++r) {
        float s = 0.0f;
        if (lane < 16) {
            for (int t = 0; t < 32; ++t)
                s += g_rows[(r * 32 + t) * DOUT + n0 + lane];
        }
        acc[r] = s;
    }
#endif

    // ---- clip + store: D vgpr r / lanes 0-15 => virtual batch r ----
    if (lane < 16) {
#pragma unroll
        for (int r = 0; r < NVB; ++r) {
            float v = acc[r];
            v = fminf(fmaxf(v, 0.0f), 1.0f);
            int side = r >> 2;
            int bi   = r & 3;
            out[(size_t)side * (size_t)B * DOUT
                + (size_t)(bbase + bi) * DOUT + (n0 + lane)] = v;
        }
    }
}

// ---------------------------------------------------------------------------
// Launch
// ---------------------------------------------------------------------------
extern "C" void kernel_launch(void* const* d_in, const int* in_sizes, int n_in,
                              void* d_out, int out_size, void* d_ws, size_t ws_size,
                              hipStream_t stream) {
    const int*   values = (const int*)  d_in[0];
    const int*   kings  = (const int*)  d_in[2];
    const float* pieces = (const float*)d_in[3];
    const float* ranks  = (const float*)d_in[4];
    const float* files  = (const float*)d_in[5];
    const float* noking = (const float*)d_in[6];
    const float* tiles  = (const float*)d_in[7];
    const float* fmask  = (const float*)d_in[8];
    float* wtab = (float*)d_ws;          // 3072*128 f32 = 1.5 MB scratch
    float* out  = (float*)d_out;

    int B = in_sizes[1];                 // 16384 (lengths count)
    int L = in_sizes[0] / B;             // 32

    build_table_kernel<<<NKB * KPC * 64, DOUT, 0, stream>>>(
        pieces, ranks, files, noking, tiles, fmask, wtab);

    size_t shmem = (size_t)ROWS * ROW_BYTES + (size_t)ROWS * sizeof(unsigned);
    accum_kernel<<<B / BATCHES_PER_BLOCK, 256, shmem, stream>>>(
        values, kings, wtab, out, B, L);
}